// INFORMForAttacker_22909355557426
// MI455X (gfx1250) — compile-verified
//
#include <hip/hip_runtime.h>

typedef __attribute__((ext_vector_type(2))) float v2f;
typedef __attribute__((ext_vector_type(8))) float v8f;

#define FDIM 256            // feature dimension of x
#define EDGES_PER_WAVE 16   // one 16x16 WMMA tile of edge-pair dot products
#define WAVES_PER_BLOCK 8
#define EDGES_PER_BLOCK (EDGES_PER_WAVE * WAVES_PER_BLOCK)

// bias = (1/nnz) * sum_e vals[e] * dot(x[rows[e]], x[cols[e]])
// Each wave handles 16 edges. A[m,k] = x[rows[m]][k], B[k,n] = x[cols[n]][k];
// D = A*B accumulated over K=256 in chunks of 4 via v_wmma_f32_16x16x4_f32.
// diag(D) are the per-edge dot products.
__global__ __launch_bounds__(256) void trace_quadform_wmma_kernel(
    const float* __restrict__ x,
    const int*   __restrict__ rows,
    const int*   __restrict__ cols,
    const float* __restrict__ vals,
    float*       __restrict__ out,
    int nnz)
{
    const int lane = threadIdx.x & 31;
    const int wave = threadIdx.x >> 5;
    const int e0   = blockIdx.x * EDGES_PER_BLOCK + wave * EDGES_PER_WAVE;

    // WMMA 16x16x4 f32 operand layout (wave32):
    //  A (16x4): lane l holds (A[l&15][koff], A[l&15][koff+1]) with koff = (l>>4)*2
    //  B (4x16): lane l holds (B[koff][l&15], B[koff+1][l&15]) — same pattern with
    //            matrix column l&15, so both are one contiguous float2 per lane.
    const int m    = lane & 15;         // edge-in-tile (A row / B column)
    const int koff = (lane >> 4) << 1;  // 0 or 2

    const int  e     = e0 + m;
    const bool valid = (e < nnz);
    const int  ec    = valid ? e : (nnz - 1);   // clamp loads; mask via scale

    const int   r     = rows[ec];
    const int   c     = cols[ec];
    const float scale = valid ? vals[ec] : 0.0f;

    const float* pa = x + (size_t)r * FDIM + koff;
    const float* pb = x + (size_t)c * FDIM + koff;

    v8f acc = {};
#pragma unroll 4
    for (int kk = 0; kk < FDIM; kk += 4) {
        v2f a = *(const v2f*)(pa + kk);
        v2f b = *(const v2f*)(pb + kk);
        // 8 args: (neg_a, A, neg_b, B, c_mod, C, reuse_a, reuse_b)
        acc = __builtin_amdgcn_wmma_f32_16x16x4_f32(
            false, a, false, b, (short)0, acc, false, false);
    }

    // Diagonal extraction from the 16x16 f32 D layout:
    //   D[m][n]: m<8  -> VGPR m,   lane n
    //            m>=8 -> VGPR m-8, lane n+16
    //   So D[m][m]: lanes 0..7 have edge m=lane in acc[lane];
    //               lanes 24..31 have edge m=lane-16 in acc[lane-24].
    int sel = -1;
    if (lane < 8)        sel = lane;
    else if (lane >= 24) sel = lane - 24;

    float diag = acc[0];
#pragma unroll
    for (int i = 1; i < 8; ++i) diag = (sel == i) ? acc[i] : diag;

    float contrib = (sel >= 0) ? diag * scale : 0.0f;

    // wave32 reduction
#pragma unroll
    for (int off = 16; off >= 1; off >>= 1)
        contrib += __shfl_xor(contrib, off, 32);

    __shared__ float wsum[WAVES_PER_BLOCK];
    if (lane == 0) wsum[wave] = contrib;
    __syncthreads();

    if (threadIdx.x == 0) {
        float s = 0.0f;
#pragma unroll
        for (int w = 0; w < WAVES_PER_BLOCK; ++w) s += wsum[w];
        atomicAdd(out, s * (1.0f / (float)nnz));
    }
}

extern "C" void kernel_launch(void* const* d_in, const int* in_sizes, int n_in,
                              void* d_out, int out_size, void* d_ws, size_t ws_size,
                              hipStream_t stream) {
    const float* x    = (const float*)d_in[0];
    const int*   rows = (const int*)d_in[1];
    const int*   cols = (const int*)d_in[2];
    const float* vals = (const float*)d_in[3];
    const int    nnz  = in_sizes[1];

    // Output accumulator must start at zero every call (harness does not re-poison).
    hipMemsetAsync(d_out, 0, sizeof(float), stream);

    const int grid = (nnz + EDGES_PER_BLOCK - 1) / EDGES_PER_BLOCK;
    trace_quadform_wmma_kernel<<<grid, 256, 0, stream>>>(
        x, rows, cols, vals, (float*)d_out, nnz);
}